// I2GCN_32676111188362
// MI455X (gfx1250) — compile-verified
//
#include <hip/hip_runtime.h>

// I2GCN on MI455X (gfx1250): fused WMMA fp32 implementation.
// N=8192 nodes, D=128 features, L=2 layers, k = int(0.2*N) = 1638.

#define NN 8192
#define DD 128
#define KTOP 1638
#define NEG_SLOPE 0.01f
#define EPSN 1e-8f
#define SIGMA_ 32.0f
#define TR 4  // rows per top-k workgroup (LDS: TR*32KB keys)

typedef __attribute__((ext_vector_type(2))) float v2f;
typedef __attribute__((ext_vector_type(8))) float v8f;

// V_WMMA_F32_16X16X4_F32: D(16x16 f32) = A(16x4 f32) * B(4x16 f32) + C
__device__ __forceinline__ v8f wmma4(v2f a, v2f b, v8f c) {
  return __builtin_amdgcn_wmma_f32_16x16x4_f32(false, a, false, b, (short)0, c,
                                               false, false);
}

// order-preserving float<->uint key transform (ascending)
__device__ __forceinline__ unsigned ordkey(float f) {
  unsigned u = __float_as_uint(f);
  return (u & 0x80000000u) ? ~u : (u | 0x80000000u);
}
__device__ __forceinline__ float ordval(unsigned u) {
  unsigned b = (u & 0x80000000u) ? (u & 0x7fffffffu) : ~u;
  return __uint_as_float(b);
}

// ---------------------------------------------------------------------------
// XW1 = x @ W1[l], XW2 = x @ W2[l]  (8192x128 @ 128x128), WMMA fp32.
// grid = N/16 blocks x 256 threads (8 waves); wave w owns 16-col slice w.
// ---------------------------------------------------------------------------
__global__ __launch_bounds__(256) void xw_kernel(
    const float* __restrict__ x, const float* __restrict__ W1,
    const float* __restrict__ W2, float* __restrict__ XW1,
    float* __restrict__ XW2) {
  const int wave = threadIdx.x >> 5;
  const int lane = threadIdx.x & 31;
  const int half = lane >> 4;  // selects K pair (A) / K rows (B)
  const int lid = lane & 15;
  const int m0 = blockIdx.x * 16;
  const int n0 = wave * 16;
  v8f acc1 = {};
  v8f acc2 = {};
  for (int k0 = 0; k0 < DD; k0 += 4) {
    const int ka = k0 + 2 * half;
    v2f a;
    a.x = x[(m0 + lid) * DD + ka];
    a.y = x[(m0 + lid) * DD + ka + 1];
    v2f b1;
    b1.x = W1[ka * DD + n0 + lid];
    b1.y = W1[(ka + 1) * DD + n0 + lid];
    v2f b2;
    b2.x = W2[ka * DD + n0 + lid];
    b2.y = W2[(ka + 1) * DD + n0 + lid];
    acc1 = wmma4(a, b1, acc1);
    acc2 = wmma4(a, b2, acc2);
  }
  for (int v = 0; v < 8; ++v) {
    const int r = m0 + v + 8 * half;
    const int c = n0 + lid;
    XW1[r * DD + c] = acc1[v];
    XW2[r * DD + c] = acc2[v];
  }
}

// ---------------------------------------------------------------------------
// invn[i] = 1/(||x_i|| + 1e-8)
// ---------------------------------------------------------------------------
__global__ __launch_bounds__(256) void invnorm_kernel(const float* __restrict__ x,
                                                      float* __restrict__ invn) {
  const int i = blockIdx.x * 256 + threadIdx.x;
  if (i >= NN) return;
  float ss = 0.f;
  const float4* xr = (const float4*)(x + (size_t)i * DD);
  for (int d = 0; d < DD / 4; ++d) {
    float4 v = xr[d];
    ss += v.x * v.x + v.y * v.y + v.z * v.z + v.w * v.w;
  }
  invn[i] = 1.0f / (sqrtf(ss) + EPSN);
}

// ---------------------------------------------------------------------------
// invP[i] = 1 / (sum_j exp(-(imp_i - imp_j)^2 / 32) + 1)   (prior row-norm)
// ---------------------------------------------------------------------------
__global__ __launch_bounds__(256) void prior_kernel(const float* __restrict__ imp,
                                                    float* __restrict__ invP) {
  __shared__ float simp[256];
  const int i = blockIdx.x * 256 + threadIdx.x;
  const float mi = imp[i];
  float sum = 0.f;
  for (int j0 = 0; j0 < NN; j0 += 256) {
    __syncthreads();
    simp[threadIdx.x] = imp[j0 + threadIdx.x];
    __syncthreads();
    for (int j = 0; j < 256; ++j) {
      float dd = mi - simp[j];
      sum += expf(-(dd * dd) / SIGMA_);
    }
  }
  invP[i] = 1.0f / (sum + 1.0f);  // +1 for the eye self-loop
}

// ---------------------------------------------------------------------------
// Exact per-row k-th largest of S[i,:] (cosine sims) via in-LDS radix select,
// plus invF[i] = 1/(sum of kept values + 1). TR=4 rows per workgroup so each
// pass over x (4MB, L2-resident) is amortized over 4 rows. LDS ~138KB
// (gfx1250 allows up to 320KB per workgroup).
// ---------------------------------------------------------------------------
__global__ __launch_bounds__(256) void topk_kernel(
    const float* __restrict__ x, const float* __restrict__ invn,
    float* __restrict__ thr_out, float* __restrict__ invF) {
  __shared__ unsigned keys[TR][NN];    // 128KB
  __shared__ float xi[TR][DD];         // 2KB
  __shared__ unsigned hist[TR][256];   // 4KB
  __shared__ unsigned sel[TR][2];
  __shared__ float red[256];
  const int row0 = blockIdx.x * TR;
  const int tid = threadIdx.x;

  for (int t = tid; t < TR * DD; t += 256) {
    const int r = t / DD, c = t % DD;
    xi[r][c] = x[(size_t)(row0 + r) * DD + c] * invn[row0 + r];
  }
  __syncthreads();

  // S rows: each thread computes TR dots per j, reusing one pass over x_j.
  for (int j = tid; j < NN; j += 256) {
    const float4* xj = (const float4*)(x + (size_t)j * DD);
    float d0 = 0.f, d1 = 0.f, d2 = 0.f, d3 = 0.f;
    for (int d = 0; d < DD / 4; ++d) {
      const float4 v = xj[d];
      const float4 a0 = *(const float4*)(&xi[0][d * 4]);
      const float4 a1 = *(const float4*)(&xi[1][d * 4]);
      const float4 a2 = *(const float4*)(&xi[2][d * 4]);
      const float4 a3 = *(const float4*)(&xi[3][d * 4]);
      d0 += a0.x * v.x + a0.y * v.y + a0.z * v.z + a0.w * v.w;
      d1 += a1.x * v.x + a1.y * v.y + a1.z * v.z + a1.w * v.w;
      d2 += a2.x * v.x + a2.y * v.y + a2.z * v.z + a2.w * v.w;
      d3 += a3.x * v.x + a3.y * v.y + a3.z * v.z + a3.w * v.w;
    }
    const float jn = invn[j];
    keys[0][j] = ordkey(d0 * jn);
    keys[1][j] = ordkey(d1 * jn);
    keys[2][j] = ordkey(d2 * jn);
    keys[3][j] = ordkey(d3 * jn);
  }
  __syncthreads();

  // MSB-first radix select of rank-KTOP-from-top, all TR rows in parallel.
  unsigned prefix[TR] = {0u, 0u, 0u, 0u};
  unsigned pmask = 0u;
  int remaining[TR] = {KTOP, KTOP, KTOP, KTOP};
  for (int pass = 0; pass < 4; ++pass) {
    const int shift = 24 - 8 * pass;
    for (int t = tid; t < TR * 256; t += 256) hist[t >> 8][t & 255] = 0u;
    __syncthreads();
    for (int j = tid; j < NN; j += 256) {
#pragma unroll
      for (int r = 0; r < TR; ++r) {
        unsigned u = keys[r][j];
        if ((u & pmask) == prefix[r])
          atomicAdd(&hist[r][(u >> shift) & 255u], 1u);
      }
    }
    __syncthreads();
    if (tid < TR) {
      const int r = tid;
      int cum = 0, b = 255;
      for (; b > 0; --b) {
        cum += (int)hist[r][b];
        if (cum >= remaining[r]) break;
      }
      if (cum < remaining[r]) cum += (int)hist[r][0];
      sel[r][0] = (unsigned)b;
      sel[r][1] = (unsigned)(remaining[r] - (cum - (int)hist[r][b]));
    }
    __syncthreads();
#pragma unroll
    for (int r = 0; r < TR; ++r) {
      prefix[r] |= sel[r][0] << shift;
      remaining[r] = (int)sel[r][1];
    }
    pmask |= 255u << shift;
    __syncthreads();
  }

  float thrv[TR], sums[TR];
#pragma unroll
  for (int r = 0; r < TR; ++r) {
    thrv[r] = ordval(prefix[r]);
    sums[r] = 0.f;
  }
  for (int j = tid; j < NN; j += 256) {
#pragma unroll
    for (int r = 0; r < TR; ++r) {
      float v = ordval(keys[r][j]);
      if (v >= thrv[r]) sums[r] += v;
    }
  }
  for (int r = 0; r < TR; ++r) {
    red[tid] = sums[r];
    __syncthreads();
    for (int s = 128; s > 0; s >>= 1) {
      if (tid < s) red[tid] += red[tid + s];
      __syncthreads();
    }
    if (tid == 0) {
      thr_out[row0 + r] = thrv[r];
      invF[row0 + r] = 1.0f / (red[0] + 1.0f);  // +1 for eye
    }
    __syncthreads();
  }
}

// ---------------------------------------------------------------------------
// Fused GCN layer: xout = lrelu( A_p @ XW1 + A_f @ XW2 ), adjacency tiles
// regenerated on the fly. Workgroup: 32 output rows x 128 cols, 8 waves.
// Per 32-wide j step: waves 0-3 build the 32x32 S tile with WMMA (this IS the
// xn@xn^T GEMM), waves 4-7 build the 32x32 P tile (exp, pre-scaled by invP);
// then all 8 waves accumulate P@XW1 + S@XW2 with WMMA.
// ---------------------------------------------------------------------------
__global__ __launch_bounds__(256) void fused_kernel(
    const float* __restrict__ x, const float* __restrict__ invn,
    const float* __restrict__ imp, const float* __restrict__ invP,
    const float* __restrict__ thr, const float* __restrict__ invF,
    const float* __restrict__ XW1, const float* __restrict__ XW2,
    float* __restrict__ xout) {
  __shared__ float xnI[32][DD];   // 16KB  normalized i rows
  __shared__ float xnJ[32][DD];   // 16KB  normalized j rows
  __shared__ float Stile[32][32]; // 4KB
  __shared__ float Ptile[32][32]; // 4KB
  __shared__ float s_impI[32], s_thr[32], s_invF[32], s_invP[32], s_impJ[32];

  const int tid = threadIdx.x;
  const int wave = tid >> 5;
  const int lane = tid & 31;
  const int half = lane >> 4;
  const int lid = lane & 15;
  const int i0 = blockIdx.x * 32;
  const int n0 = wave * 16;

  for (int t = tid; t < 32 * DD; t += 256) {
    const int r = t / DD, c = t % DD;
    xnI[r][c] = x[(size_t)(i0 + r) * DD + c] * invn[i0 + r];
  }
  if (tid < 32) {
    s_impI[tid] = imp[i0 + tid];
    s_thr[tid] = thr[i0 + tid];
    s_invF[tid] = invF[i0 + tid];
    s_invP[tid] = invP[i0 + tid];
  }

  v8f acc0 = {};  // rows i0..i0+15, cols n0..n0+15
  v8f acc1 = {};  // rows i0+16..i0+31

  for (int j0 = 0; j0 < NN; j0 += 32) {
    __syncthreads();  // prior phase B done with tiles (also covers init stage)
    for (int t = tid; t < 32 * DD; t += 256) {
      const int r = t / DD, c = t % DD;
      xnJ[r][c] = x[(size_t)(j0 + r) * DD + c] * invn[j0 + r];
    }
    if (tid < 32) s_impJ[tid] = imp[j0 + tid];
    // prefetch next j slab (x rows + XW slabs) into cache
    if (j0 + 32 < NN) {
      __builtin_prefetch(x + (size_t)(j0 + 32 + tid / 8) * DD, 0, 3);
      __builtin_prefetch(XW1 + (size_t)(j0 + 32 + tid / 8) * DD, 0, 3);
      __builtin_prefetch(XW2 + (size_t)(j0 + 32 + tid / 8) * DD, 0, 3);
    }
    __syncthreads();

    if (wave < 4) {
      // S 16x16 sub-tile (mt,nt) = xnI(16x128) @ xnJ(128x16)^T
      const int mt = wave & 1, nt = wave >> 1;
      v8f sa = {};
      for (int k0 = 0; k0 < DD; k0 += 4) {
        const int ka = k0 + 2 * half;
        v2f a;
        a.x = xnI[mt * 16 + lid][ka];
        a.y = xnI[mt * 16 + lid][ka + 1];
        v2f b;  // B[k][n] = xnJ[n][k]
        b.x = xnJ[nt * 16 + lid][ka];
        b.y = xnJ[nt * 16 + lid][ka + 1];
        sa = wmma4(a, b, sa);
      }
      for (int v = 0; v < 8; ++v)
        Stile[mt * 16 + v + 8 * half][nt * 16 + lid] = sa[v];
    } else {
      // P tile: (exp(-(di-dj)^2/32) + eye) * invP[row]
      const int w = wave - 4;
      for (int e = 0; e < 8; ++e) {
        const int idx = (w * 32 + lane) * 8 + e;  // 0..1023
        const int r = idx >> 5, c = idx & 31;
        float dd = s_impI[r] - s_impJ[c];
        float p = expf(-(dd * dd) / SIGMA_);
        if (i0 + r == j0 + c) p += 1.0f;
        Ptile[r][c] = p * s_invP[r];
      }
    }
    __syncthreads();
    // mask + eye + row-normalize S tile
    for (int e = tid; e < 32 * 32; e += 256) {
      const int r = e >> 5, c = e & 31;
      float s = Stile[r][c];
      float v = (s >= s_thr[r]) ? s : 0.0f;
      if (i0 + r == j0 + c) v += 1.0f;
      Stile[r][c] = v * s_invF[r];
    }
    __syncthreads();
    // phase B: acc += P' @ XW1 + S' @ XW2 over this 32-deep K slab
    for (int k0 = 0; k0 < 32; k0 += 4) {
      const int ka = k0 + 2 * half;
      v2f b1;
      b1.x = XW1[(size_t)(j0 + ka) * DD + n0 + lid];
      b1.y = XW1[(size_t)(j0 + ka + 1) * DD + n0 + lid];
      v2f b2;
      b2.x = XW2[(size_t)(j0 + ka) * DD + n0 + lid];
      b2.y = XW2[(size_t)(j0 + ka + 1) * DD + n0 + lid];
      v2f p0, s0, p1, s1;
      p0.x = Ptile[lid][ka];      p0.y = Ptile[lid][ka + 1];
      s0.x = Stile[lid][ka];      s0.y = Stile[lid][ka + 1];
      p1.x = Ptile[16 + lid][ka]; p1.y = Ptile[16 + lid][ka + 1];
      s1.x = Stile[16 + lid][ka]; s1.y = Stile[16 + lid][ka + 1];
      acc0 = wmma4(p0, b1, acc0);
      acc0 = wmma4(s0, b2, acc0);
      acc1 = wmma4(p1, b1, acc1);
      acc1 = wmma4(s1, b2, acc1);
    }
  }
  // leaky relu + store
  for (int v = 0; v < 8; ++v) {
    const int r = i0 + v + 8 * half;
    const int c = n0 + lid;
    float f0 = acc0[v];
    float f1 = acc1[v];
    f0 = (f0 >= 0.f) ? f0 : NEG_SLOPE * f0;
    f1 = (f1 >= 0.f) ? f1 : NEG_SLOPE * f1;
    xout[(size_t)r * DD + c] = f0;
    xout[(size_t)(r + 16) * DD + c] = f1;
  }
}

// ---------------------------------------------------------------------------
// out = x_final + h * importances   (h = input to last layer = layer-0 output)
// ---------------------------------------------------------------------------
__global__ __launch_bounds__(256) void resid_kernel(
    const float* __restrict__ xB, const float* __restrict__ xA,
    const float* __restrict__ imp, float* __restrict__ out) {
  const int t = blockIdx.x * 256 + threadIdx.x;
  if (t >= NN * DD) return;
  out[t] = xB[t] + xA[t] * imp[t / DD];
}

extern "C" void kernel_launch(void* const* d_in, const int* in_sizes, int n_in,
                              void* d_out, int out_size, void* d_ws,
                              size_t ws_size, hipStream_t stream) {
  const float* x = (const float*)d_in[0];     // [N, D]
  const float* imp = (const float*)d_in[1];   // [N, 1]
  const float* W1 = (const float*)d_in[2];    // [2, D, D]
  const float* W2 = (const float*)d_in[3];    // [2, D, D]
  float* out = (float*)d_out;

  float* ws = (float*)d_ws;
  float* XW1 = ws;
  float* XW2 = XW1 + (size_t)NN * DD;
  float* xA = XW2 + (size_t)NN * DD;
  float* xB = xA + (size_t)NN * DD;
  float* invn = xB + (size_t)NN * DD;
  float* thr = invn + NN;
  float* invF = thr + NN;
  float* invP = invF + NN;

  prior_kernel<<<NN / 256, 256, 0, stream>>>(imp, invP);

  const float* xcur = x;
  float* outs[2] = {xA, xB};
  for (int l = 0; l < 2; ++l) {
    invnorm_kernel<<<NN / 256, 256, 0, stream>>>(xcur, invn);
    xw_kernel<<<NN / 16, 256, 0, stream>>>(xcur, W1 + (size_t)l * DD * DD,
                                           W2 + (size_t)l * DD * DD, XW1, XW2);
    topk_kernel<<<NN / TR, 256, 0, stream>>>(xcur, invn, thr, invF);
    fused_kernel<<<NN / 32, 256, 0, stream>>>(xcur, invn, imp, invP, thr, invF,
                                              XW1, XW2, outs[l]);
    xcur = outs[l];
  }
  resid_kernel<<<(NN * DD) / 256, 256, 0, stream>>>(xB, xA, imp, out);
}